// TMLoRA_28587302322946
// MI455X (gfx1250) — compile-verified
//
#include <hip/hip_runtime.h>
#include <math.h>

// ---------------------------------------------------------------------------
// Fused TM-LoRA forward for MI455X (gfx1250, wave32).
// Memory-bound problem (~512 MiB moved, ~2.1 GFLOP): single pass over x,
// single pass over out. Both GEMMs use V_WMMA_F32_16X16X4_F32 (K=4 matches
// the LoRA rank exactly; router+A projection accumulates K=4096 in steps of 4,
// split across the 8 waves of the block and reduced through LDS).
// Staging uses GLOBAL_LOAD_ASYNC_TO_LDS_B128 when the toolchain exposes it
// (ASYNCcnt-tracked direct-to-LDS DMA), else coalesced NT b128 loads.
// ---------------------------------------------------------------------------

typedef float v2f __attribute__((ext_vector_type(2)));
typedef float v4f __attribute__((ext_vector_type(4)));
typedef float v8f __attribute__((ext_vector_type(8)));
typedef int   i4v __attribute__((vector_size(4 * sizeof(int))));

typedef __attribute__((address_space(1))) void* gvoidp;
typedef __attribute__((address_space(3))) void* lvoidp;
typedef __attribute__((address_space(1))) i4v*  gi4p;
typedef __attribute__((address_space(3))) i4v*  li4p;

#define D_IN     4096
#define D_OUT    4096
#define N_EXP    8
#define RANK     4
#define NWROWS   12              // 8 router rows + 4 A rows (rows 12..15 zero)
#define TOP_K    4
#define SCALING  8.0f            // alpha / rank = 32/4

#define MTOK     16              // tokens per block (WMMA M)
#define KC       512             // K-chunk staged in LDS
#define PAD      4               // row pad (floats) -> conflict-free ds_load_b64
#define LDW      (KC + PAD)      // 516 (stride 516 words => banks 4*lane)
#define NWAVES   8
#define THREADS  (NWAVES * 32)

#if __has_builtin(__builtin_amdgcn_global_load_async_to_lds_b128) && \
    __has_builtin(__builtin_amdgcn_s_wait_asynccnt)
#define HAS_ASYNC_LDS 1
#else
#define HAS_ASYNC_LDS 0
#endif

// Copy 16 bytes global -> LDS (per lane). Async DMA path when available.
__device__ __forceinline__ void copy16_to_lds(const float* g, float* l) {
#if HAS_ASYNC_LDS
    __builtin_amdgcn_global_load_async_to_lds_b128(
        (gi4p)(gvoidp)(void*)g,
        (li4p)(lvoidp)l, 0, 0);
#else
    *(v4f*)l = __builtin_nontemporal_load((const v4f*)g);
#endif
}

__device__ __forceinline__ void staging_fence_then_barrier() {
#if HAS_ASYNC_LDS
    __builtin_amdgcn_s_wait_asynccnt(0);   // barrier does NOT wait on ASYNCcnt
#endif
    __syncthreads();
}

__global__ __launch_bounds__(THREADS)
void tmlora_fused(const float* __restrict__ x,
                  const float* __restrict__ Aw,     // [RANK, D_IN]
                  const float* __restrict__ Bw,     // [D_OUT, RANK]
                  const float* __restrict__ Rw,     // [N_EXP, D_IN]
                  const float* __restrict__ ev,     // [N_EXP, RANK]
                  float* __restrict__ out)          // [T, D_OUT]
{
    __shared__ float xs[MTOK * LDW];        // x tile      (33 KB)
    __shared__ float ws[16 * LDW];          // weight tile (33 KB, rows 12..15 = 0)
    __shared__ float red[NWAVES][THREADS];  // partial-C reduction (8 KB)
    __shared__ float scores_s[16][16];      // reduced C tile
    __shared__ float h_s[MTOK][RANK];       // gelu(hidden) * SCALING

    const int tid   = threadIdx.x;
    const int wave  = tid >> 5;
    const int lane  = tid & 31;
    const int t0    = blockIdx.x * MTOK;
    const int mrow  = lane & 15;            // A-frag token row / B-frag column
    const int khalf = (lane >> 4) << 1;     // K sub-pair: lanes 0-15 -> K0/K1, 16-31 -> K2/K3

    // Zero the padding weight rows (12..15) once.
    for (int i = tid; i < 4 * LDW; i += THREADS)
        ws[NWROWS * LDW + i] = 0.0f;

    // ---------------- Phase A: scores/A-dot via WMMA over K ----------------
    v8f cacc = {};
    for (int k0 = 0; k0 < D_IN; k0 += KC) {
        // Stage x tile: 16 rows x KC floats, 16 B per lane per step.
        for (int i = tid; i < MTOK * (KC / 4); i += THREADS) {
            const int m  = i / (KC / 4);
            const int k4 = (i % (KC / 4)) * 4;
            copy16_to_lds(x + (size_t)(t0 + m) * D_IN + k0 + k4,
                          xs + m * LDW + k4);
        }
        // Stage combined weights: router rows 0..7, A rows 8..11.
        for (int i = tid; i < NWROWS * (KC / 4); i += THREADS) {
            const int m  = i / (KC / 4);
            const int k4 = (i % (KC / 4)) * 4;
            const float* src = (m < N_EXP) ? (Rw + (size_t)m * D_IN)
                                           : (Aw + (size_t)(m - N_EXP) * D_IN);
            copy16_to_lds(src + k0 + k4, ws + m * LDW + k4);
        }
        // Prefetch next x chunk into L2 while we compute this one.
        if (k0 + KC < D_IN) {
            const float* nx = x + (size_t)(t0 + (tid & 15)) * D_IN
                                + k0 + KC + ((tid >> 4) * 32);
            __builtin_prefetch(nx, 0, 1);
        }
        staging_fence_then_barrier();

        // Each wave owns a 64-wide K slice of this chunk: 16 WMMA steps of K=4.
        const int kwbase = wave * (KC / NWAVES);
        #pragma unroll
        for (int s = 0; s < (KC / NWAVES) / 4; ++s) {
            const int kk = kwbase + s * 4 + khalf;
            v2f a = *(const v2f*)(xs + mrow * LDW + kk);   // A: x[m][k..k+1]
            v2f b = *(const v2f*)(ws + mrow * LDW + kk);   // B: W[n][k..k+1]
            cacc = __builtin_amdgcn_wmma_f32_16x16x4_f32(
                false, a, false, b, (short)0, cacc, false, false);
        }
        __syncthreads();
    }

    // Reduce the 8 per-wave partial C tiles through LDS.
    #pragma unroll
    for (int v = 0; v < 8; ++v)
        red[wave][v * 32 + lane] = cacc[v];
    __syncthreads();
    {
        float ssum = 0.0f;
        #pragma unroll
        for (int w = 0; w < NWAVES; ++w) ssum += red[w][tid];
        const int v = tid >> 5, l = tid & 31;
        const int m = v + ((l >> 4) << 3);   // C layout: lanes 16-31 hold M=v+8
        const int n = l & 15;
        scores_s[m][n] = ssum;
    }
    __syncthreads();

    // ---------------- Phase B: top-4 -> softmax -> mix -> GELU -------------
    if (tid < MTOK) {
        float sc[N_EXP];
        #pragma unroll
        for (int e = 0; e < N_EXP; ++e) sc[e] = scores_s[tid][e];
        float tv[TOP_K]; int ti[TOP_K];
        #pragma unroll
        for (int k = 0; k < TOP_K; ++k) {
            float best = -INFINITY; int bi = 0;
            #pragma unroll
            for (int e = 0; e < N_EXP; ++e)
                if (sc[e] > best) { best = sc[e]; bi = e; }
            tv[k] = best; ti[k] = bi; sc[bi] = -INFINITY;
        }
        const float mx = tv[0];
        float wk[TOP_K], wsum = 0.0f;
        #pragma unroll
        for (int k = 0; k < TOP_K; ++k) { wk[k] = __expf(tv[k] - mx); wsum += wk[k]; }
        const float inv = 1.0f / wsum;
        #pragma unroll
        for (int r = 0; r < RANK; ++r) {
            float et = 0.0f;
            #pragma unroll
            for (int k = 0; k < TOP_K; ++k)
                et += wk[k] * inv * ev[ti[k] * RANK + r];
            const float hv = scores_s[tid][N_EXP + r] + et;
            const float g  = 0.5f * hv * (1.0f + erff(hv * 0.70710678118654752f));
            h_s[tid][r] = g * SCALING;   // fold alpha/rank scaling here
        }
    }
    __syncthreads();

    // ---------------- Phase C: out = h @ Bw^T, one WMMA per 16-col tile ----
    v2f ha;
    ha.x = h_s[mrow][khalf];
    ha.y = h_s[mrow][khalf + 1];
    const int colsPerWave = D_OUT / NWAVES;         // 512
    for (int i = 0; i < colsPerWave / 16; ++i) {    // 32 tiles per wave
        const int n0 = wave * colsPerWave + i * 16;
        // B frag: lane n supplies Bw[n0+n][khalf..khalf+1] (Bw row-major [D_OUT,4])
        v2f b = *(const v2f*)(Bw + (size_t)(n0 + mrow) * RANK + khalf);
        v8f c = {};
        c = __builtin_amdgcn_wmma_f32_16x16x4_f32(
            false, ha, false, b, (short)0, c, false, false);
        const int nst   = n0 + (lane & 15);
        const int mbase = t0 + ((lane >> 4) << 3);
        #pragma unroll
        for (int v = 0; v < 8; ++v)
            __builtin_nontemporal_store(
                c[v], out + (size_t)(mbase + v) * D_OUT + nst);
    }
}

extern "C" void kernel_launch(void* const* d_in, const int* in_sizes, int n_in,
                              void* d_out, int out_size, void* d_ws, size_t ws_size,
                              hipStream_t stream) {
    const float* x  = (const float*)d_in[0];  // [B,S,D_in]
    const float* Aw = (const float*)d_in[1];  // [RANK, D_in]
    const float* Bw = (const float*)d_in[2];  // [D_out, RANK]
    const float* Rw = (const float*)d_in[3];  // [N_EXP, D_in]
    const float* ev = (const float*)d_in[4];  // [N_EXP, RANK]
    float* out = (float*)d_out;

    const int ntok = in_sizes[0] / D_IN;      // B*S = 16384
    dim3 grid(ntok / MTOK);                   // 1024 blocks, 8 waves each
    tmlora_fused<<<grid, THREADS, 0, stream>>>(x, Aw, Bw, Rw, ev, out);
}